// MPNN_52012053955020
// MI455X (gfx1250) — compile-verified
//
#include <hip/hip_runtime.h>

// ---------------------------------------------------------------------------
// 2-layer GCN for MI455X (gfx1250, wave32).
//   layer(x) = relu( (scatter_add(x[src] -> dst) + x) @ W + b )
// Scatter is L2-atomic bound (x fits in 192MB L2); GEMM is bandwidth-bound,
// done with V_WMMA_F32_16X16X4_F32 (f32 in/out, no precision loss).
// ---------------------------------------------------------------------------

typedef float v2f __attribute__((ext_vector_type(2)));
typedef float v8f __attribute__((ext_vector_type(8)));

#define DIM 64          // feature dimension (compile-time per reference)
#define LDS_W_STRIDE 80 // 64 + 16 pad: keeps the two half-wave B reads in
                        // disjoint LDS bank groups (banks shift by 32)

// ---- self-loop init: agg = x (float4 vectorized copy) ---------------------
__global__ void gcn_selfloop_copy(const float4* __restrict__ in,
                                  float4* __restrict__ out, int n4) {
  int i = blockIdx.x * blockDim.x + threadIdx.x;
  if (i < n4) out[i] = in[i];
}

// ---- edge scatter: agg[dst[e]] += x[src[e]] -------------------------------
// 16 threads per edge; each thread moves one float4 (coalesced 256B gather of
// the source row) and issues 4 global_atomic_add_f32 into the dst row.
__global__ void gcn_scatter_add(const float* __restrict__ x,
                                const long long* __restrict__ src,
                                const long long* __restrict__ dst,
                                float* __restrict__ agg, int nE) {
  unsigned t = blockIdx.x * blockDim.x + threadIdx.x;
  unsigned e = t >> 4;
  if (e >= (unsigned)nE) return;
  unsigned q = t & 15;  // which float4 of the 64-float row

  long long s = src[e];
  long long d = dst[e];

  const float4 v = *(const float4*)(x + (size_t)s * DIM + q * 4);
  float* p = agg + (size_t)d * DIM + q * 4;
  atomicAdd(p + 0, v.x);
  atomicAdd(p + 1, v.y);
  atomicAdd(p + 2, v.z);
  atomicAdd(p + 3, v.w);
}

// ---- out = relu(A @ W + b), A: N x 64 (N % 32 == 0), W: 64 x 64 -----------
// 256 threads = 8 waves; wave w handles row-tile (w>>2) of 2, col-tile (w&3)
// of 4 -> 32x64 output per block. Each wave chains 16 x V_WMMA_F32_16X16X4_F32.
//
// Operand layouts (ISA 7.12.2, wave32):
//   A 16x4:  lane L -> M = L&15; regs {0,1} hold K = 2*(L>>4) + {0,1}
//   B 4x16:  lane L -> N = L&15; regs {0,1} hold K = 2*(L>>4) + {0,1}
//   C/D:     lane L -> N = L&15; reg r holds M = r + 8*(L>>4)
__global__ void gcn_gemm_relu_wmma(const float* __restrict__ A,
                                   const float* __restrict__ W,
                                   const float* __restrict__ b,
                                   float* __restrict__ out, int N) {
  __shared__ float sW[DIM * LDS_W_STRIDE];

  // Cooperative stage of W (4096 floats = 1024 float4) into padded LDS.
  for (int i = threadIdx.x; i < DIM * (DIM / 4); i += blockDim.x) {
    int row = i >> 4;   // / (DIM/4 floats4 per row)
    int c4  = i & 15;
    float4 w = ((const float4*)W)[i];
    *(float4*)(&sW[row * LDS_W_STRIDE + c4 * 4]) = w;
  }
  __syncthreads();

  const int wave = threadIdx.x >> 5;
  const int lane = threadIdx.x & 31;
  const int half = lane >> 4;  // selects K pair (A/B) and M group (C/D)
  const int lrow = lane & 15;  // M for A, N for B/C/D

  const int m0 = (blockIdx.x * 2 + (wave >> 2)) * 16;
  const int n0 = (wave & 3) * 16;
  if (m0 >= N) return;  // wave-uniform: EXEC stays all-1s for WMMA

  const float* Arow = A + (size_t)(m0 + lrow) * DIM;

  v8f acc = {};
#pragma unroll
  for (int k0 = 0; k0 < DIM; k0 += 4) {
    const int k = k0 + half * 2;

    v2f a;
    {
      float2 av = *(const float2*)(Arow + k);  // 8B aligned: k is even
      a.x = av.x;
      a.y = av.y;
    }
    v2f bb;
    bb.x = sW[(k + 0) * LDS_W_STRIDE + n0 + lrow];
    bb.y = sW[(k + 1) * LDS_W_STRIDE + n0 + lrow];

    // (neg_a, A, neg_b, B, c_mod, C, reuse_a, reuse_b)
    acc = __builtin_amdgcn_wmma_f32_16x16x4_f32(
        false, a, false, bb, (short)0, acc, false, false);
  }

  const float bias = b[n0 + lrow];
#pragma unroll
  for (int r = 0; r < 8; ++r) {
    const int row = m0 + r + half * 8;
    float v = acc[r] + bias;
    out[(size_t)row * DIM + n0 + lrow] = v > 0.0f ? v : 0.0f;
  }
}

// ---------------------------------------------------------------------------
extern "C" void kernel_launch(void* const* d_in, const int* in_sizes, int n_in,
                              void* d_out, int out_size, void* d_ws, size_t ws_size,
                              hipStream_t stream) {
  const float* x      = (const float*)d_in[0];
  const long long* ei = (const long long*)d_in[1];  // int64 edge_index (2, E)
  const float* W1     = (const float*)d_in[2];
  const float* b1     = (const float*)d_in[3];
  const float* W2     = (const float*)d_in[4];
  const float* b2     = (const float*)d_in[5];
  float* out          = (float*)d_out;

  const int Dm = in_sizes[3];        // 64
  const int N  = in_sizes[0] / Dm;   // 100000
  const int E  = in_sizes[1] / 2;    // 1000000
  const long long* src = ei;
  const long long* dst = ei + E;

  float* agg = (float*)d_ws;                 // N*D floats
  float* h   = agg + (size_t)N * Dm;         // N*D floats

  const int n4         = (N * Dm) / 4;
  const int copyBlocks = (n4 + 255) / 256;
  const int scatBlocks = (E * 16 + 255) / 256;
  const int gemmBlocks = N / 32;             // N % 32 == 0 for this workload

  // ---- layer 1: h = relu((scatter(x) + x) @ W1 + b1) ----
  gcn_selfloop_copy<<<copyBlocks, 256, 0, stream>>>((const float4*)x,
                                                    (float4*)agg, n4);
  gcn_scatter_add<<<scatBlocks, 256, 0, stream>>>(x, src, dst, agg, E);
  gcn_gemm_relu_wmma<<<gemmBlocks, 256, 0, stream>>>(agg, W1, b1, h, N);

  // ---- layer 2: out = relu((scatter(h) + h) @ W2 + b2) ----
  gcn_selfloop_copy<<<copyBlocks, 256, 0, stream>>>((const float4*)h,
                                                    (float4*)agg, n4);
  gcn_scatter_add<<<scatBlocks, 256, 0, stream>>>(h, src, dst, agg, E);
  gcn_gemm_relu_wmma<<<gemmBlocks, 256, 0, stream>>>(agg, W2, b2, out, N);
}